// MOELoraLinear_20469814133118
// MI455X (gfx1250) — compile-verified
//
#include <hip/hip_runtime.h>

typedef float v2f __attribute__((ext_vector_type(2)));
typedef float v8f __attribute__((ext_vector_type(8)));

#define B_  4
#define S_  2048
#define D_  4096
#define O_  4096
#define E_  8
#define R_  8
#define M_  (B_ * S_)     // 8192
#define ER_ (E_ * R_)     // 64
#define SCALING_ 2.0f     // 16 / 8
#define KB_ 16            // k-block per LDS stage
#define LDST 20           // padded LDS row stride (floats): conflict-free

// ---- gfx1250 async global->LDS path (guarded; falls back to sync loads) ----
#if __has_builtin(__builtin_amdgcn_global_load_async_to_lds_b128) && \
    __has_builtin(__builtin_amdgcn_s_wait_asynccnt)
#define USE_ASYNC 1
typedef __attribute__((__vector_size__(16))) int v4i_t;
__device__ __forceinline__ void async_b128(void* lds, const void* gm) {
    __builtin_amdgcn_global_load_async_to_lds_b128(
        (__attribute__((address_space(1))) v4i_t*)gm,
        (__attribute__((address_space(3))) v4i_t*)lds,
        0, 0);
}
#define WAIT_ASYNC(n) __builtin_amdgcn_s_wait_asynccnt(n)
#else
#define USE_ASYNC 0
#define WAIT_ASYNC(n)
#endif

// ---------------------------------------------------------------------------
// Kernel 1: mean-pool x over the sequence dim -> xm[B, D]
// ---------------------------------------------------------------------------
__global__ __launch_bounds__(256) void k_mean(const float* __restrict__ x,
                                              float* __restrict__ xm) {
    int d = blockIdx.x * 256 + threadIdx.x;
    int b = blockIdx.y;
    const float* p = x + (size_t)b * S_ * D_ + d;
    float s = 0.f;
    for (int i = 0; i < S_; ++i) s += p[(size_t)i * D_];
    xm[b * D_ + d] = s * (1.0f / (float)S_);
}

// ---------------------------------------------------------------------------
// Kernel 2: router logits[b,e] = dot(xm[b], router_W[e]) + router_b[e]
// ---------------------------------------------------------------------------
__global__ __launch_bounds__(256) void k_logits(const float* __restrict__ xm,
                                                const float* __restrict__ rW,
                                                const float* __restrict__ rb,
                                                float* __restrict__ logits) {
    int p = blockIdx.x;
    int b = p >> 3, e = p & 7;
    __shared__ float red[256];
    float s = 0.f;
    for (int d = threadIdx.x; d < D_; d += 256)
        s += xm[b * D_ + d] * rW[e * D_ + d];
    red[threadIdx.x] = s;
    __syncthreads();
    for (int off = 128; off > 0; off >>= 1) {
        if (threadIdx.x < off) red[threadIdx.x] += red[threadIdx.x + off];
        __syncthreads();
    }
    if (threadIdx.x == 0) logits[p] = red[0] + rb[e];
}

// ---------------------------------------------------------------------------
// Kernel 3: softmax over E=8 per batch (ROUTER_TEMP == 1)
// ---------------------------------------------------------------------------
__global__ void k_softmax(const float* __restrict__ logits,
                          float* __restrict__ w) {
    int t = threadIdx.x;
    if (t < B_) {
        float l[E_];
        float mx = -1e30f;
        for (int e = 0; e < E_; ++e) { l[e] = logits[t * E_ + e]; mx = fmaxf(mx, l[e]); }
        float s = 0.f;
        for (int e = 0; e < E_; ++e) { l[e] = expf(l[e] - mx); s += l[e]; }
        float inv = 1.0f / s;
        for (int e = 0; e < E_; ++e) w[t * E_ + e] = l[e] * inv;
    }
}

// ---------------------------------------------------------------------------
// Kernel 4: hw[m, er] = SCALING * weights[b(m), e] * (x @ lora_A^T)
//   M-tile 128/block, N=64, K=4096. 8 waves, each 16x64 (4 accumulators).
//   Double-buffered LDS stages, async global->LDS when available.
// ---------------------------------------------------------------------------
__global__ __launch_bounds__(256) void k_hw(const float* __restrict__ x,
                                            const float* __restrict__ lA,   // [ER_, D_]
                                            const float* __restrict__ w,    // [B_, E_]
                                            float* __restrict__ hw) {       // [M_, ER_]
    __shared__ __align__(16) float As[2][128][LDST];
    __shared__ __align__(16) float Bs[2][64][LDST];

    const int tid  = threadIdx.x;
    const int wave = tid >> 5;
    const int lane = tid & 31;
    const int l15  = lane & 15;
    const int ksel = (lane >> 4) * 2;
    const int m0   = blockIdx.x * 128;

    v8f acc[4] = {};

    // A loader: 2 threads/row, 8 floats each (2x b128)
    const int arow  = tid >> 1;
    const int ahalf = (tid & 1) * 8;
    const float* aptr = x + (size_t)(m0 + arow) * D_ + ahalf;
    // B loader: 4 threads/row, 4 floats each (1x b128)
    const int brow = tid >> 2;
    const int bq   = (tid & 3) * 4;
    const float* bptr = lA + (size_t)brow * D_ + bq;

    auto stage = [&](int buf, int kb) {
#if USE_ASYNC
        async_b128(&As[buf][arow][ahalf],     aptr + kb);
        async_b128(&As[buf][arow][ahalf + 4], aptr + kb + 4);
        async_b128(&Bs[buf][brow][bq],        bptr + kb);
#else
        *(float4*)&As[buf][arow][ahalf]     = *(const float4*)(aptr + kb);
        *(float4*)&As[buf][arow][ahalf + 4] = *(const float4*)(aptr + kb + 4);
        *(float4*)&Bs[buf][brow][bq]        = *(const float4*)(bptr + kb);
        __builtin_prefetch(aptr + kb + KB_, 0, 3);
#endif
    };

    auto compute = [&](int buf) {
        const int rA = wave * 16 + l15;
#pragma unroll
        for (int kk = 0; kk < KB_; kk += 4) {
            v2f a;
            a.x = As[buf][rA][kk + ksel];
            a.y = As[buf][rA][kk + ksel + 1];
#pragma unroll
            for (int j = 0; j < 4; ++j) {
                int nB = j * 16 + l15;
                v2f bf;
                bf.x = Bs[buf][nB][kk + ksel];
                bf.y = Bs[buf][nB][kk + ksel + 1];
                acc[j] = __builtin_amdgcn_wmma_f32_16x16x4_f32(
                    false, a, false, bf, (short)0, acc[j], false, false);
            }
        }
    };

    stage(0, 0);
    int cur = 0, kb = 0;
    for (; kb + KB_ < D_; kb += KB_) {
        stage(cur ^ 1, kb + KB_);
        WAIT_ASYNC(3);                 // retire previous stage's 3 async ops
        __syncthreads();
        compute(cur);
        __syncthreads();
        cur ^= 1;
    }
    WAIT_ASYNC(0);
    __syncthreads();
    compute(cur);

    // store, folding router weight * SCALING
    const int mbase = m0 + wave * 16 + ((lane >> 4) * 8);
#pragma unroll
    for (int j = 0; j < 4; ++j) {
        int n = j * 16 + l15;
        int e = n >> 3;
#pragma unroll
        for (int v = 0; v < 8; ++v) {
            int m = mbase + v;
            int b = m >> 11;                       // m / S_
            float ws = w[b * E_ + e] * SCALING_;
            hw[(size_t)m * ER_ + n] = acc[j][v] * ws;
        }
    }
}

// ---------------------------------------------------------------------------
// Kernel 5: fused out = x @ W_base^T + bias + hw @ Btilde
//   Block tile 128(M) x 256(N), 8 waves (2x4), each wave 64x64 = 4x4 WMMA
//   tiles. K loop runs D_ + 64; the last 4 stages feed the LoRA rank-64
//   GEMM (A <- hw, B <- lora_B viewed as [er, o]). Double-buffered LDS.
// ---------------------------------------------------------------------------
__global__ __launch_bounds__(256) void k_main(const float* __restrict__ x,
                                              const float* __restrict__ Wb,   // [O_, D_]
                                              const float* __restrict__ bias, // [O_]
                                              const float* __restrict__ lB,   // [E_, O_, R_]
                                              const float* __restrict__ hw,   // [M_, ER_]
                                              float* __restrict__ out) {      // [M_, O_]
    __shared__ __align__(16) float As[2][128][LDST];   // 20 KB
    __shared__ __align__(16) float Bs[2][256][LDST];   // 40 KB  (Bs[n][k])

    const int tid  = threadIdx.x;
    const int wave = tid >> 5;
    const int lane = tid & 31;
    const int l15  = lane & 15;
    const int ksel = (lane >> 4) * 2;
    const int wm   = wave & 1;     // 2 waves along M
    const int wn   = wave >> 1;    // 4 waves along N
    const int m0   = blockIdx.x * 128;
    const int n0   = blockIdx.y * 256;

    v8f acc[4][4] = {};

    // A loader: 2 threads/row, 8 floats each (2x b128)
    const int arow  = tid >> 1;
    const int ahalf = (tid & 1) * 8;
    const float* xrow = x  + (size_t)(m0 + arow) * D_  + ahalf;
    const float* hrow = hw + (size_t)(m0 + arow) * ER_ + ahalf;
    // B loader: 1 thread/row, 16 floats (4x b128)
    const int nglob = n0 + tid;
    const float* wrow = Wb + (size_t)nglob * D_;

    auto stage = [&](int buf, int kb) {
        if (kb < D_) {
#if USE_ASYNC
            async_b128(&As[buf][arow][ahalf],     xrow + kb);
            async_b128(&As[buf][arow][ahalf + 4], xrow + kb + 4);
#pragma unroll
            for (int j = 0; j < 16; j += 4)
                async_b128(&Bs[buf][tid][j], wrow + kb + j);
#else
            *(float4*)&As[buf][arow][ahalf]     = *(const float4*)(xrow + kb);
            *(float4*)&As[buf][arow][ahalf + 4] = *(const float4*)(xrow + kb + 4);
#pragma unroll
            for (int j = 0; j < 16; j += 4)
                *(float4*)&Bs[buf][tid][j] = *(const float4*)(wrow + kb + j);
            __builtin_prefetch(xrow + kb + KB_, 0, 3);
            __builtin_prefetch(wrow + kb + KB_, 0, 3);
#endif
        } else {
            const int kbL = kb - D_;              // 0, 16, 32, 48
#if USE_ASYNC
            async_b128(&As[buf][arow][ahalf],     hrow + kbL);
            async_b128(&As[buf][arow][ahalf + 4], hrow + kbL + 4);
#pragma unroll
            for (int j = 0; j < 16; j += 4) {
                int kk = kbL + j;                 // e = kk>>3, r = kk&7
                async_b128(&Bs[buf][tid][j],
                           lB + ((size_t)(kk >> 3) * O_ + nglob) * R_ + (kk & 7));
            }
#else
            *(float4*)&As[buf][arow][ahalf]     = *(const float4*)(hrow + kbL);
            *(float4*)&As[buf][arow][ahalf + 4] = *(const float4*)(hrow + kbL + 4);
#pragma unroll
            for (int j = 0; j < 16; j += 4) {
                int kk = kbL + j;
                *(float4*)&Bs[buf][tid][j] =
                    *(const float4*)(lB + ((size_t)(kk >> 3) * O_ + nglob) * R_ + (kk & 7));
            }
#endif
        }
    };

    auto compute = [&](int buf) {
#pragma unroll
        for (int kk = 0; kk < KB_; kk += 4) {
            v2f a[4], b[4];
#pragma unroll
            for (int i = 0; i < 4; ++i) {
                const float* p = &As[buf][wm * 64 + i * 16 + l15][kk + ksel];
                a[i].x = p[0]; a[i].y = p[1];
            }
#pragma unroll
            for (int j = 0; j < 4; ++j) {
                const float* p = &Bs[buf][wn * 64 + j * 16 + l15][kk + ksel];
                b[j].x = p[0]; b[j].y = p[1];
            }
#pragma unroll
            for (int i = 0; i < 4; ++i)
#pragma unroll
                for (int j = 0; j < 4; ++j)
                    acc[i][j] = __builtin_amdgcn_wmma_f32_16x16x4_f32(
                        false, a[i], false, b[j], (short)0, acc[i][j], false, false);
        }
    };

    const int Ktot = D_ + ER_;
    stage(0, 0);
    int cur = 0, kb = 0;
    for (; kb + KB_ < Ktot; kb += KB_) {
        stage(cur ^ 1, kb + KB_);
        WAIT_ASYNC(6);                 // retire previous stage's 6 async ops
        __syncthreads();
        compute(cur);
        __syncthreads();
        cur ^= 1;
    }
    WAIT_ASYNC(0);
    __syncthreads();
    compute(cur);

    // epilogue: + bias, store
    const int mbase = m0 + wm * 64 + ((lane >> 4) * 8);
#pragma unroll
    for (int i = 0; i < 4; ++i)
#pragma unroll
        for (int j = 0; j < 4; ++j) {
            int n = n0 + wn * 64 + j * 16 + l15;
            float bb = bias[n];
#pragma unroll
            for (int v = 0; v < 8; ++v) {
                int m = mbase + i * 16 + v;
                out[(size_t)m * O_ + n] = acc[i][j][v] + bb;
            }
        }
}

// ---------------------------------------------------------------------------
// Launch
// ---------------------------------------------------------------------------
extern "C" void kernel_launch(void* const* d_in, const int* in_sizes, int n_in,
                              void* d_out, int out_size, void* d_ws, size_t ws_size,
                              hipStream_t stream) {
    const float* x        = (const float*)d_in[0];
    const float* W_base   = (const float*)d_in[1];
    const float* b_base   = (const float*)d_in[2];
    const float* lora_A   = (const float*)d_in[3];   // [E,R,D] == [ER, D] row-major
    const float* lora_B   = (const float*)d_in[4];   // [E,O,R]
    const float* router_W = (const float*)d_in[5];
    const float* router_b = (const float*)d_in[6];
    float* out = (float*)d_out;

    float* ws      = (float*)d_ws;
    float* xm      = ws;                         // B*D = 16384
    float* logits  = ws + (size_t)B_ * D_;       // 32
    float* weights = logits + 32;                // 32
    float* hw      = weights + 32;               // M*ER = 524288 (16B aligned)

    k_mean   <<<dim3(D_ / 256, B_), 256, 0, stream>>>(x, xm);
    k_logits <<<B_ * E_, 256, 0, stream>>>(xm, router_W, router_b, logits);
    k_softmax<<<1, 32, 0, stream>>>(logits, weights);
    k_hw     <<<M_ / 128, 256, 0, stream>>>(x, lora_A, weights, hw);
    k_main   <<<dim3(M_ / 128, O_ / 256), 256, 0, stream>>>(x, W_base, b_base,
                                                            lora_B, hw, out);
}